// SAGE_15479062135292
// MI455X (gfx1250) — compile-verified
//
#include <hip/hip_runtime.h>

#define N_NODES 100000
#define N_EDGES 1600000
#define DHID 128
#define DOUT3 40

typedef __attribute__((ext_vector_type(2))) float v2f;
typedef __attribute__((ext_vector_type(8))) float v8f;

__global__ void zero_kernel(float* __restrict__ p, int n) {
  int t = blockIdx.x * blockDim.x + threadIdx.x;
  if (t < n) p[t] = 0.0f;
}

__global__ void deg_kernel(const int* __restrict__ dst, float* __restrict__ deg, int e) {
  int t = blockIdx.x * blockDim.x + threadIdx.x;
  if (t < e) atomicAdd(&deg[dst[t]], 1.0f);
}

__global__ void invdeg_kernel(float* __restrict__ deg, int n) {
  int t = blockIdx.x * blockDim.x + threadIdx.x;
  if (t < n) deg[t] = 1.0f / fmaxf(deg[t], 1.0f);
}

// Out[M x Dout] = A[M x Din] @ W[Din x Dout] (+ bias), fp32 WMMA 16x16x4.
// One wave per 16x16 output tile; 4 waves (4 M-tiles) per block.
__global__ void gemm_wmma_kernel(const float* __restrict__ A,
                                 const float* __restrict__ W,
                                 const float* __restrict__ bias,
                                 float* __restrict__ Out,
                                 int M, int Din, int Dout) {
  const int wave  = threadIdx.x >> 5;
  const int lane  = threadIdx.x & 31;
  const int mtile = blockIdx.x * 4 + wave;
  const int m0    = mtile << 4;
  if (m0 >= M) return;                       // wave-uniform: EXEC all-1s for WMMA waves
  const int hh    = lane >> 4;               // 0 -> K pair {0,1}; 1 -> K pair {2,3}
  const int l16   = lane & 15;
  const int arow  = m0 + l16;                // A-matrix: row striped in-lane
  const int bcol0 = (blockIdx.y << 4) + l16; // B/C/D: row striped across lanes
  const bool colok = bcol0 < Dout;
  const int bcol  = colok ? bcol0 : 0;       // clamp (mask on store)

  v8f c = {};
  const float* Arow = A + (size_t)arow * Din;
#pragma unroll 4
  for (int k = 0; k < Din; k += 4) {
    const int ka = k + hh * 2;
    const float2 av = *reinterpret_cast<const float2*>(Arow + ka);
    v2f a, b;
    a[0] = av.x;
    a[1] = av.y;
    b[0] = W[(size_t)ka * Dout + bcol];
    b[1] = W[(size_t)(ka + 1) * Dout + bcol];
    c = __builtin_amdgcn_wmma_f32_16x16x4_f32(false, a, false, b, (short)0, c,
                                              false, false);
  }
  if (colok) {
    const float bv = (bias != nullptr) ? bias[bcol] : 0.0f;
#pragma unroll
    for (int r = 0; r < 8; ++r) {
      const int row = m0 + hh * 8 + r;       // C/D layout: VGPR r, half-wave selects M+8
      Out[(size_t)row * Dout + bcol] = c[r] + bv;
    }
  }
}

// acc[dst] += msg[src], D = 4*DQ channels; one thread = one (edge, 4-channel quad)
template <int DQ>
__global__ void scatter_kernel(const float4* __restrict__ msg,
                               const int* __restrict__ src,
                               const int* __restrict__ dst,
                               float* __restrict__ acc,
                               unsigned nthreads) {
  unsigned t = blockIdx.x * blockDim.x + threadIdx.x;
  if (t >= nthreads) return;
  unsigned e = t / DQ;
  unsigned q = t - e * DQ;
  const int s = src[e];
  const int d = dst[e];
  const float4 v = msg[(size_t)s * DQ + q];
  float* a = acc + ((size_t)d * DQ + q) * 4;
  atomicAdd(a + 0, v.x);
  atomicAdd(a + 1, v.y);
  atomicAdd(a + 2, v.z);
  atomicAdd(a + 3, v.w);
}

template <int D>
__global__ void finalize_kernel(const float* __restrict__ acc,
                                const float* __restrict__ tmpR,
                                const float* __restrict__ rdeg,
                                float* __restrict__ out,
                                int total, int dorelu) {
  int t = blockIdx.x * blockDim.x + threadIdx.x;
  if (t >= total) return;
  int i = t / D;
  float v = fmaf(acc[t], rdeg[i], tmpR[t]);
  out[t] = dorelu ? fmaxf(v, 0.0f) : v;
}

static inline int cdiv(long a, long b) { return (int)((a + b - 1) / b); }

extern "C" void kernel_launch(void* const* d_in, const int* in_sizes, int n_in,
                              void* d_out, int out_size, void* d_ws, size_t ws_size,
                              hipStream_t stream) {
  const float* x    = (const float*)d_in[0];
  const int*   ei   = (const int*)d_in[1];
  const int*   src  = ei;
  const int*   dstp = ei + N_EDGES;
  const float* Wl1 = (const float*)d_in[2];
  const float* bl1 = (const float*)d_in[3];
  const float* Wr1 = (const float*)d_in[4];
  const float* Wl2 = (const float*)d_in[5];
  const float* bl2 = (const float*)d_in[6];
  const float* Wr2 = (const float*)d_in[7];
  const float* Wl3 = (const float*)d_in[8];
  const float* bl3 = (const float*)d_in[9];
  const float* Wr3 = (const float*)d_in[10];
  float* out = (float*)d_out;

  float* ws = (float*)d_ws;
  const size_t NF = (size_t)N_NODES * DHID;   // 12.8M floats
  float* bufA = ws;                            // tmpL / hidden ping
  float* bufB = ws + NF;                       // tmpL / hidden pong
  float* tmpR = ws + 2 * NF;                   // root-path GEMM result
  float* acc  = ws + 3 * NF;                   // scatter accumulator
  float* rdeg = ws + 4 * NF;                   // 1/deg

  const int Mtiles = (N_NODES + 15) / 16;      // 6250 exact
  dim3 blk(128);
  dim3 grid8(cdiv(Mtiles, 4), DHID / 16);      // Dout = 128
  dim3 grid3(cdiv(Mtiles, 4), cdiv(DOUT3, 16));// Dout = 40 -> 3 N-tiles (masked)

  // ---- degrees (once, reused by all layers) ----
  zero_kernel<<<cdiv(N_NODES, 256), 256, 0, stream>>>(rdeg, N_NODES);
  deg_kernel<<<cdiv(N_EDGES, 256), 256, 0, stream>>>(dstp, rdeg, N_EDGES);
  invdeg_kernel<<<cdiv(N_NODES, 256), 256, 0, stream>>>(rdeg, N_NODES);

  // ---- layer 1: SAGEConv(128 -> 128) + ReLU ----
  gemm_wmma_kernel<<<grid8, blk, 0, stream>>>(x, Wl1, nullptr, bufA, N_NODES, DHID, DHID);
  gemm_wmma_kernel<<<grid8, blk, 0, stream>>>(x, Wr1, bl1, tmpR, N_NODES, DHID, DHID);
  zero_kernel<<<cdiv((long)NF, 256), 256, 0, stream>>>(acc, (int)NF);
  scatter_kernel<32><<<cdiv((long)N_EDGES * 32, 256), 256, 0, stream>>>(
      (const float4*)bufA, src, dstp, acc, (unsigned)N_EDGES * 32u);
  finalize_kernel<DHID><<<cdiv((long)NF, 256), 256, 0, stream>>>(
      acc, tmpR, rdeg, bufA, (int)NF, 1);

  // ---- layer 2: SAGEConv(128 -> 128) + ReLU ----
  gemm_wmma_kernel<<<grid8, blk, 0, stream>>>(bufA, Wl2, nullptr, bufB, N_NODES, DHID, DHID);
  gemm_wmma_kernel<<<grid8, blk, 0, stream>>>(bufA, Wr2, bl2, tmpR, N_NODES, DHID, DHID);
  zero_kernel<<<cdiv((long)NF, 256), 256, 0, stream>>>(acc, (int)NF);
  scatter_kernel<32><<<cdiv((long)N_EDGES * 32, 256), 256, 0, stream>>>(
      (const float4*)bufB, src, dstp, acc, (unsigned)N_EDGES * 32u);
  finalize_kernel<DHID><<<cdiv((long)NF, 256), 256, 0, stream>>>(
      acc, tmpR, rdeg, bufB, (int)NF, 1);

  // ---- layer 3: SAGEConv(128 -> 40), no activation ----
  const int NC = N_NODES * DOUT3;
  gemm_wmma_kernel<<<grid3, blk, 0, stream>>>(bufB, Wl3, nullptr, bufA, N_NODES, DHID, DOUT3);
  gemm_wmma_kernel<<<grid3, blk, 0, stream>>>(bufB, Wr3, bl3, tmpR, N_NODES, DHID, DOUT3);
  zero_kernel<<<cdiv(NC, 256), 256, 0, stream>>>(acc, NC);
  scatter_kernel<10><<<cdiv((long)N_EDGES * 10, 256), 256, 0, stream>>>(
      (const float4*)bufA, src, dstp, acc, (unsigned)N_EDGES * 10u);
  finalize_kernel<DOUT3><<<cdiv(NC, 256), 256, 0, stream>>>(
      acc, tmpR, rdeg, out, NC, 0);
}